// MVPartSegmentation_77653008711905
// MI455X (gfx1250) — compile-verified
//
#include <hip/hip_runtime.h>

typedef __attribute__((ext_vector_type(2))) float v2f;
typedef __attribute__((ext_vector_type(4))) float v4f;
typedef __attribute__((ext_vector_type(8))) float v8f;

#define NUM_CLASSES 16
#define D_HID 12            // hidden channels per head
#define P_OUT 7             // output logits per head
#define CH_IN 21            // backbone channels (20 via WMMA + 1 via rank-1 update)
#define NS 32               // bs*views
#define VIEWS 8
#define HW (128 * 128)
#define BN_EPS 1e-5f

#define WAVES_PER_BLOCK 8
#define TILES_PER_WAVE 4
#define CHUNKS 32           // CHUNKS * WAVES_PER_BLOCK * TILES_PER_WAVE * 16 == HW

// out[n,p,pix] = relu( W2[k] @ relu( BN( W1[k] @ x[n,:,pix] + b1 ) ) + b2 ),  k = cls[n/VIEWS]
// (exact: mask-then-max over 16 classes == relu of the selected class's logits)
__global__ __launch_bounds__(256)
void mvpartseg_head_kernel(const float* __restrict__ feats_in,
                           const int*   __restrict__ cls,
                           const float* __restrict__ W1,
                           const float* __restrict__ b1,
                           const float* __restrict__ gamma1,
                           const float* __restrict__ beta1,
                           const float* __restrict__ rm1,
                           const float* __restrict__ rv1,
                           const float* __restrict__ W2,
                           const float* __restrict__ b2,
                           float* __restrict__ out,        // (NS, P_OUT, HW)
                           float* __restrict__ feats_out)  // (NS, CH_IN, HW) = copy of input
{
    // per-wave scratch tile: h1 in [col][d] layout (bank-conflict-free, 16B aligned rows)
    __shared__ float lds[WAVES_PER_BLOCK][16 * D_HID];

    const int tid  = threadIdx.x;
    const int wv   = tid >> 5;
    const int lane = tid & 31;
    const int hi   = lane >> 4;      // half-wave select (K/M split per ISA layouts)
    const int col  = lane & 15;

    const int n = blockIdx.y;                 // flattened (batch, view) sample
    const int k = cls[n / VIEWS];             // selected class head (wave-uniform)

    // ---------------- wave-uniform A matrices in registers ----------------
    // Layer-1 A: 16x4 f32 tiles over c = 0..19. lane%16 = M(=d, padded 12->16);
    // K(c) = 4*kk + 2*hi + vgpr  (dense: no bounds checks needed)
    v2f a1[5];
#pragma unroll
    for (int kk = 0; kk < 5; ++kk) {
#pragma unroll
        for (int v = 0; v < 2; ++v) {
            const int c = kk * 4 + 2 * hi + v;            // 0..19
            a1[kk][v] = (col < D_HID) ? W1[(k * D_HID + col) * CH_IN + c] : 0.0f;
        }
    }
    // c = 20 handled as rank-1 update in accumulator layout: d = v + 8*hi
    float w20[8];
#pragma unroll
    for (int v = 0; v < 8; ++v) {
        const int d = v + 8 * hi;
        w20[v] = (d < D_HID) ? W1[(k * D_HID + d) * CH_IN + 20] : 0.0f;
    }
    // Layer-2 A: M = p (padded 7->16), K = d (12 exact = 3 steps of 4)
    v2f a2[3];
#pragma unroll
    for (int kk = 0; kk < 3; ++kk) {
#pragma unroll
        for (int v = 0; v < 2; ++v) {
            const int d = kk * 4 + 2 * hi + v;            // always < 12
            a2[kk][v] = (col < P_OUT) ? W2[(k * P_OUT + col) * D_HID + d] : 0.0f;
        }
    }
    // Folded BN (+conv bias) for accumulator layout: VGPR v -> d = v + 8*hi
    float scale[8], shift[8];
#pragma unroll
    for (int v = 0; v < 8; ++v) {
        const int d = v + 8 * hi;
        if (d < D_HID) {
            const float inv = gamma1[k * D_HID + d] / sqrtf(rv1[k * D_HID + d] + BN_EPS);
            scale[v] = inv;
            shift[v] = b1[k * D_HID + d] * inv + beta1[k * D_HID + d] - rm1[k * D_HID + d] * inv;
        } else {
            scale[v] = 0.0f;   // padding rows stay zero
            shift[v] = 0.0f;
        }
    }
    float bias2[P_OUT];
#pragma unroll
    for (int v = 0; v < P_OUT; ++v) bias2[v] = b2[k * P_OUT + v];

    const float* fin  = feats_in  + (size_t)n * CH_IN * HW;
    float*       fout = feats_out + (size_t)n * CH_IN * HW;
    float*       oout = out       + (size_t)n * P_OUT * HW;

#pragma unroll 1
    for (int t = 0; t < TILES_PER_WAVE; ++t) {
        const int tile = (blockIdx.x * WAVES_PER_BLOCK + wv) * TILES_PER_WAVE + t;
        const int pix0 = tile * 16;

        // -------- layer 1: acc1 = W1[:,0:20] @ x (5 dense f32 WMMAs, no divergence) ----
        v8f acc1 = {};
#pragma unroll
        for (int kk = 0; kk < 5; ++kk) {
            v2f bmat;
#pragma unroll
            for (int v = 0; v < 2; ++v) {
                const int c = kk * 4 + 2 * hi + v;        // B row (K); col = pixel
                const int idx = c * HW + pix0 + col;
                const float x = fin[idx];
                fout[idx] = x;    // fused feats copy: each element loaded exactly once
                bmat[v] = x;
            }
            acc1 = __builtin_amdgcn_wmma_f32_16x16x4_f32(
                false, a1[kk], false, bmat, (short)0, acc1, false, false);
        }
        // rank-1 update for c = 20 (all lanes load; halves read identical addresses)
        const int idx20 = 20 * HW + pix0 + col;
        const float x20 = fin[idx20];
        if (hi == 0) fout[idx20] = x20;                   // copy row 20 once
#pragma unroll
        for (int v = 0; v < 8; ++v) acc1[v] += w20[v] * x20;

        // -------- BN + ReLU; stash h1 tile to LDS in [col][d] layout ----------------
        float h[8];
#pragma unroll
        for (int v = 0; v < 8; ++v)
            h[v] = fmaxf(acc1[v] * scale[v] + shift[v], 0.0f);
        // d = 8*hi + 0..3 : always valid -> one b128 store per lane
        *(v4f*)&lds[wv][col * D_HID + 8 * hi] = (v4f){h[0], h[1], h[2], h[3]};
        // d = 4..7 only exist for hi==0 -> single-branch b128 store
        if (hi == 0)
            *(v4f*)&lds[wv][col * D_HID + 4] = (v4f){h[4], h[5], h[6], h[7]};
        __builtin_amdgcn_wave_barrier();
        asm volatile("s_wait_dscnt 0" ::: "memory");      // wave-local LDS RAW

        // -------- layer 2: acc2 = W2 @ h1 (3 chained f32 WMMAs, K = 12 exact) --------
        v8f acc2 = {};
#pragma unroll
        for (int kk = 0; kk < 3; ++kk) {
            const v2f bmat = *(const v2f*)&lds[wv][col * D_HID + kk * 4 + 2 * hi];
            acc2 = __builtin_amdgcn_wmma_f32_16x16x4_f32(
                false, a2[kk], false, bmat, (short)0, acc2, false, false);
        }
        __builtin_amdgcn_wave_barrier();

        // -------- mask+max over classes == relu(selected logits); coalesced store ----
        if (hi == 0) {                                    // p = v lives in lanes 0-15
#pragma unroll
            for (int v = 0; v < P_OUT; ++v)
                oout[v * HW + pix0 + col] = fmaxf(acc2[v] + bias2[v], 0.0f);
        }
    }
}

extern "C" void kernel_launch(void* const* d_in, const int* in_sizes, int n_in,
                              void* d_out, int out_size, void* d_ws, size_t ws_size,
                              hipStream_t stream) {
    (void)in_sizes; (void)n_in; (void)out_size; (void)d_ws; (void)ws_size;

    const float* mvimages = (const float*)d_in[0];   // (4,8,21,128,128)
    const int*   cls      = (const int*)  d_in[1];   // (4,1)
    const float* W1       = (const float*)d_in[2];   // (16,12,21)
    const float* b1       = (const float*)d_in[3];   // (16,12)
    const float* gamma1   = (const float*)d_in[4];   // (16,12)
    const float* beta1    = (const float*)d_in[5];   // (16,12)
    const float* rm1      = (const float*)d_in[6];   // (16,12)
    const float* rv1      = (const float*)d_in[7];   // (16,12)
    const float* W2       = (const float*)d_in[8];   // (16,7,12)
    const float* b2       = (const float*)d_in[9];   // (16,7)

    float* out       = (float*)d_out;                           // (32,7,128,128)
    float* feats_out = (float*)d_out + (size_t)NS * P_OUT * HW; // (32,21,128,128)

    dim3 grid(CHUNKS, NS);       // 32 pixel-chunks x 32 samples
    dim3 block(256);             // 8 waves; each wave: 4 tiles of 16 pixels
    mvpartseg_head_kernel<<<grid, block, 0, stream>>>(
        mvimages, cls, W1, b1, gamma1, beta1, rm1, rv1, W2, b2, out, feats_out);
}